// MeanAggLayer_44719199485973
// MI455X (gfx1250) — compile-verified
//
#include <hip/hip_runtime.h>

typedef float v2f __attribute__((ext_vector_type(2)));
typedef float v8f __attribute__((ext_vector_type(8)));

#define DIM  128   // feature dim (K)
#define OUTF 128   // output dim
#define BM   128   // rows per block (8 waves x 16)
#define BN   64    // output cols per LDS split
#define LDK  132   // padded K stride of transposed W tile (bank-conflict-free b64)

// ---------------------------------------------------------------------------
// Kernel 1: h[i] = x[i] + mean_{e: dst==i} x[src[e]]
// edge_dst is sorted -> each node's edges are contiguous; one wave per node
// binary-searches [lo,hi), zero atomics. Edge sources are fetched 32-at-a-time
// by the whole wave (one coalesced load), broadcast with __shfl, and the row
// gathers are issued 4 deep for memory-level parallelism.
// ---------------------------------------------------------------------------
__global__ __launch_bounds__(256) void agg_mean_kernel(
    const float* __restrict__ x,
    const int* __restrict__ esrc,
    const int* __restrict__ edst,
    float* __restrict__ h, int n, int e) {
  const int wave = blockIdx.x * (blockDim.x >> 5) + (threadIdx.x >> 5);
  const int lane = threadIdx.x & 31;
  if (wave >= n) return;

  // lower_bound(edge_dst, wave)
  int lo;
  {
    int l = 0, r = e;
    while (l < r) { int m = (l + r) >> 1; if (edst[m] < wave) l = m + 1; else r = m; }
    lo = l;
  }
  // upper_bound(edge_dst, wave)
  int hi;
  {
    int l = lo, r = e;
    while (l < r) { int m = (l + r) >> 1; if (edst[m] <= wave) l = m + 1; else r = m; }
    hi = l;
  }

  float4 acc = make_float4(0.f, 0.f, 0.f, 0.f);
  const int foff = lane * 4;

  for (int base = lo; base < hi; base += 32) {
    const int m = min(hi - base, 32);
    // one coalesced vector load of up to 32 edge sources
    const int s = (lane < m) ? esrc[base + lane] : 0;

    int j = 0;
    for (; j + 4 <= m; j += 4) {   // 4 independent row-gathers in flight
      const int s0 = __shfl(s, j + 0, 32);
      const int s1 = __shfl(s, j + 1, 32);
      const int s2 = __shfl(s, j + 2, 32);
      const int s3 = __shfl(s, j + 3, 32);
      const float4 v0 = *(const float4*)(x + (size_t)s0 * DIM + foff);
      const float4 v1 = *(const float4*)(x + (size_t)s1 * DIM + foff);
      const float4 v2 = *(const float4*)(x + (size_t)s2 * DIM + foff);
      const float4 v3 = *(const float4*)(x + (size_t)s3 * DIM + foff);
      acc.x += v0.x + v1.x + v2.x + v3.x;
      acc.y += v0.y + v1.y + v2.y + v3.y;
      acc.z += v0.z + v1.z + v2.z + v3.z;
      acc.w += v0.w + v1.w + v2.w + v3.w;
    }
    for (; j < m; ++j) {
      const int sj = __shfl(s, j, 32);
      const float4 v = *(const float4*)(x + (size_t)sj * DIM + foff);
      acc.x += v.x; acc.y += v.y; acc.z += v.z; acc.w += v.w;
    }
  }

  const float inv = 1.0f / fmaxf((float)(hi - lo), 1.0f);
  const float4 xi = *(const float4*)(x + (size_t)wave * DIM + foff);
  float4 o;
  o.x = xi.x + acc.x * inv;
  o.y = xi.y + acc.y * inv;
  o.z = xi.z + acc.z * inv;
  o.w = xi.w + acc.w * inv;
  *(float4*)(h + (size_t)wave * DIM + foff) = o;
}

// ---------------------------------------------------------------------------
// Kernel 2: out = relu(h @ W + b) via V_WMMA_F32_16X16X4_F32.
// 256 threads = 8 waves; block covers BM=128 rows, each wave a 16-row band.
// A fragments stream from global (each h byte read once per split); W staged transposed
// into LDS in two 64-col splits (LDK=132 pad -> conflict-free ds_load_b64).
// Interior blocks take a uniform unguarded store path (no exec-mask churn).
// ---------------------------------------------------------------------------
__global__ __launch_bounds__(256) void wmma_gemm_kernel(
    const float* __restrict__ h,
    const float* __restrict__ W,
    const float* __restrict__ bias,
    float* __restrict__ out, int n) {
  __shared__ float Bs[BN * LDK];   // Bs[col * LDK + k] = W[k][nBase + col]

  const int tid  = threadIdx.x;
  const int wave = tid >> 5;
  const int lane = tid & 31;
  const int half = lane >> 4;      // 0: K=0..1 / M=v ; 1: K=2..3 / M=v+8
  const int l16  = lane & 15;
  const int rowBase = blockIdx.x * BM;
  const int mOff = wave * 16;
  const bool full = (rowBase + BM) <= n;   // uniform across block

  int arow = rowBase + mOff + l16;
  if (arow >= n) arow = n - 1;     // clamp reads; tail stores guarded below
  const float* __restrict__ aptr = h + (size_t)arow * DIM + half * 2;

  for (int split = 0; split < 2; ++split) {
    const int nBase = split * BN;

    // stage W[:, nBase..nBase+BN) transposed into LDS
    for (int i = tid; i < (DIM * BN) / 4; i += 256) {
      const int k  = i / (BN / 4);
      const int c4 = (i % (BN / 4)) * 4;
      const float4 wv = *(const float4*)(W + (size_t)k * OUTF + nBase + c4);
      Bs[(c4 + 0) * LDK + k] = wv.x;
      Bs[(c4 + 1) * LDK + k] = wv.y;
      Bs[(c4 + 2) * LDK + k] = wv.z;
      Bs[(c4 + 3) * LDK + k] = wv.w;
    }
    __syncthreads();

    v8f acc[BN / 16];
#pragma unroll
    for (int t = 0; t < BN / 16; ++t) {
      v8f z = {0.f, 0.f, 0.f, 0.f, 0.f, 0.f, 0.f, 0.f};
      acc[t] = z;
    }

#pragma unroll 4
    for (int k0 = 0; k0 < DIM; k0 += 4) {
      const v2f a = *(const v2f*)(aptr + k0);               // A: M=l16, K=k0+2*half..+1
#pragma unroll
      for (int t = 0; t < BN / 16; ++t) {
        const v2f bf = *(const v2f*)(&Bs[(t * 16 + l16) * LDK + k0 + half * 2]);
        acc[t] = __builtin_amdgcn_wmma_f32_16x16x4_f32(
            /*neg_a=*/false, a, /*neg_b=*/false, bf,
            /*c_mod=*/(short)0, acc[t], /*reuse_a=*/false, /*reuse_b=*/false);
      }
    }

    // epilogue: bias + relu
    if (full) {
      // uniform fast path: straight-line coalesced stores, EXEC stays all-1s
#pragma unroll
      for (int t = 0; t < BN / 16; ++t) {
        const int col = nBase + t * 16 + l16;
        const float bb = bias[col];
#pragma unroll
        for (int v = 0; v < 8; ++v) {
          const int r = rowBase + mOff + v + half * 8;      // C/D: M = v + 8*half
          out[(size_t)r * OUTF + col] = fmaxf(acc[t][v] + bb, 0.0f);
        }
      }
    } else {
#pragma unroll
      for (int t = 0; t < BN / 16; ++t) {
        const int col = nBase + t * 16 + l16;
        const float bb = bias[col];
#pragma unroll
        for (int v = 0; v < 8; ++v) {
          const int r = rowBase + mOff + v + half * 8;
          if (r < n) out[(size_t)r * OUTF + col] = fmaxf(acc[t][v] + bb, 0.0f);
        }
      }
    }
    __syncthreads();   // before next split overwrites Bs
  }
}

extern "C" void kernel_launch(void* const* d_in, const int* in_sizes, int n_in,
                              void* d_out, int out_size, void* d_ws, size_t ws_size,
                              hipStream_t stream) {
  const float* x    = (const float*)d_in[0];
  const float* W    = (const float*)d_in[1];
  const float* bias = (const float*)d_in[2];
  const int*   esrc = (const int*)d_in[3];
  const int*   edst = (const int*)d_in[4];
  float*       out  = (float*)d_out;
  float*       hbuf = (float*)d_ws;           // N*DIM f32 scratch (~51 MB)

  const int n = in_sizes[0] / DIM;
  const int e = in_sizes[3];

  dim3 blk(256);
  dim3 g1((unsigned)((n + 7) / 8));           // 8 waves per block, 1 wave per node
  agg_mean_kernel<<<g1, blk, 0, stream>>>(x, esrc, edst, hbuf, n, e);

  dim3 g2((unsigned)((n + BM - 1) / BM));
  wmma_gemm_kernel<<<g2, blk, 0, stream>>>(hbuf, W, bias, out, n);
}